// TtLlamaAttention_optimized_30545807409344
// MI455X (gfx1250) — compile-verified
//
#include <hip/hip_runtime.h>
#include <math.h>

// ---- problem constants (from the reference) -------------------------------
#define DIMX     4096
#define QKVN     6144          // (32 + 2*8) * 128
#define NHEADS   32
#define NKV      8
#define HD       128
#define BATCHSZ  32
#define CACHELEN 2176          // 2048 + 128

typedef __attribute__((ext_vector_type(2))) float v2f;
typedef __attribute__((ext_vector_type(4))) float v4f;
typedef __attribute__((ext_vector_type(8))) float v8f;

// ---------------------------------------------------------------------------
// GEMM: C(32 x N) = A(32 x K) @ B(K x N), fp32 WMMA 16x16x4.
// grid.x = N/16; block = 256 (8 waves). Each wave: both 16-row tiles of one
// 16-column strip, over a private K/8 chunk; partials reduced via LDS.
// Software-pipelined: fragments for step i+1 are loaded before the WMMAs of
// step i, all addressing via pointer increments (no per-iter u64 multiplies).
// Fragment layouts per CDNA5 ISA 7.12.2:
//   A 16x4 : lanes 0-15 -> {K0,K1} in v0,v1 ; lanes 16-31 -> {K2,K3}
//   B 4x16 : vgpr j, lanes 0-15 -> row K=j ; lanes 16-31 -> row K=j+2
//   C 16x16: vgpr j, lanes 0-15 -> M=j, N=lane ; lanes 16-31 -> M=j+8
// ---------------------------------------------------------------------------
__global__ __launch_bounds__(256)
void gemm32_wmma(const float* __restrict__ A, const float* __restrict__ B,
                 float* __restrict__ C, int K, int N) {
  __shared__ float red[2][8][32][8];   // tile, wave, lane, vgpr  (16 KB)
  const int tid  = threadIdx.x;
  const int wid  = tid >> 5;
  const int lane = tid & 31;
  const int half = lane >> 4;
  const int l15  = lane & 15;
  const int n0   = blockIdx.x * 16;

  const int kchunk = K >> 3;           // K split across 8 waves
  const int kbeg   = wid * kchunk;
  const int iters  = kchunk >> 2;      // 4 K-steps per WMMA pair
  const size_t nstep = (size_t)N * 4;  // 4 B-rows per iteration

  v8f acc0 = {};
  v8f acc1 = {};

  // lane-resident streaming pointers
  const float* a0p = A + (size_t)l15 * K        + kbeg + 2 * half;
  const float* a1p = A + (size_t)(16 + l15) * K + kbeg + 2 * half;
  const float* b0p = B + (size_t)(kbeg + 2 * half) * N + n0 + l15;  // row ko
  const float* b1p = b0p + N;                                       // row ko+1

  // prologue: fragments for step 0
  v2f a0 = *(const v2f*)a0p;
  v2f a1 = *(const v2f*)a1p;
  v2f bf;
  bf.x = *b0p;
  bf.y = *b1p;
  a0p += 4; a1p += 4; b0p += nstep; b1p += nstep;

  for (int i = 1; i < iters; ++i) {
    // prefetch B cachelines 4 rows (1 iteration) ahead of the loads
    __builtin_prefetch(b0p + nstep, 0, 1);
    __builtin_prefetch(b1p + nstep, 0, 1);
    // load step-i fragments while step-(i-1) WMMAs execute
    v2f na0 = *(const v2f*)a0p;
    v2f na1 = *(const v2f*)a1p;
    v2f nbf;
    nbf.x = *b0p;
    nbf.y = *b1p;
    a0p += 4; a1p += 4; b0p += nstep; b1p += nstep;

    acc0 = __builtin_amdgcn_wmma_f32_16x16x4_f32(false, a0, false, bf,
                                                 (short)0, acc0, false, false);
    acc1 = __builtin_amdgcn_wmma_f32_16x16x4_f32(false, a1, false, bf,
                                                 (short)0, acc1, false, false);
    a0 = na0; a1 = na1; bf = nbf;
  }
  // epilogue: last step
  acc0 = __builtin_amdgcn_wmma_f32_16x16x4_f32(false, a0, false, bf,
                                               (short)0, acc0, false, false);
  acc1 = __builtin_amdgcn_wmma_f32_16x16x4_f32(false, a1, false, bf,
                                               (short)0, acc1, false, false);

#pragma unroll
  for (int j = 0; j < 8; ++j) {
    red[0][wid][lane][j] = acc0[j];
    red[1][wid][lane][j] = acc1[j];
  }
  __syncthreads();

  // 512 output elements per block; sum the 8 wave partials and scatter.
  for (int slot = tid; slot < 512; slot += 256) {
    const int t   = slot >> 8;
    const int rem = slot & 255;
    const int l   = rem >> 3;
    const int j   = rem & 7;
    float s = 0.f;
#pragma unroll
    for (int w = 0; w < 8; ++w) s += red[t][w][l][j];
    const int m = t * 16 + j + 8 * (l >> 4);
    const int n = n0 + (l & 15);
    C[(size_t)m * N + n] = s;
  }
}

// ---------------------------------------------------------------------------
// RoPE (half-split) applied in-place to Q (32 heads) and K (8 heads) rows of
// the qkv workspace, at position *sp.
// ---------------------------------------------------------------------------
__global__ __launch_bounds__(256)
void rope_kernel(float* __restrict__ qkv, const int* __restrict__ sp) {
  const int idx   = blockIdx.x * 256 + threadIdx.x;
  const int total = BATCHSZ * (NHEADS + NKV) * (HD / 2);
  if (idx >= total) return;
  const int d = idx & 63;
  const int h = (idx >> 6) % (NHEADS + NKV);
  const int b = idx / (64 * (NHEADS + NKV));
  const size_t base = (size_t)b * QKVN + (size_t)h * HD;  // q heads then k heads contiguous
  const float pos = (float)(*sp);
  const float inv = __expf(-((float)(2 * d) / (float)HD) * __logf(10000.0f));
  const float ang = pos * inv;
  const float c = __cosf(ang);
  const float s = __sinf(ang);
  const float lo = qkv[base + d];
  const float hi = qkv[base + 64 + d];
  qkv[base + d]      = lo * c - hi * s;
  qkv[base + 64 + d] = hi * c + lo * s;
}

// ---------------------------------------------------------------------------
// Flash-decode GQA: one block per (batch, kv-group); 8 waves split the
// sequence with per-wave online softmax, merged through LDS. Row start_pos
// is read from the fresh (RoPE'd) K/V in the qkv workspace (caches are not
// mutated; the reference only returns attn @ wo).
// ---------------------------------------------------------------------------
__global__ __launch_bounds__(256)
void attn_kernel(const float* __restrict__ qkv, const float* __restrict__ ck,
                 const float* __restrict__ cv, float* __restrict__ attn,
                 const int* __restrict__ sp) {
  __shared__ float qs[4 * HD];         // 4 query heads of this group (scaled)
  __shared__ float m_s[8][4];
  __shared__ float l_s[8][4];
  __shared__ float o_s[8][4][HD];      // 16 KB
  const int tid  = threadIdx.x;
  const int wid  = tid >> 5;
  const int lane = tid & 31;
  const int b = blockIdx.x >> 3;
  const int g = blockIdx.x & 7;
  const int start_pos = *sp;
  const float scale = 0.08838834764831845f;  // 1/sqrt(128)

  const float* qrow = qkv + (size_t)b * QKVN + (size_t)g * 4 * HD;
  for (int i = tid; i < 4 * HD; i += 256) qs[i] = qrow[i] * scale;
  __syncthreads();

  float mi[4], li[4];
  v4f   oi[4];
#pragma unroll
  for (int h = 0; h < 4; ++h) { mi[h] = -3.0e38f; li[h] = 0.f; oi[h] = (v4f){0.f, 0.f, 0.f, 0.f}; }

  const float* knew = qkv + (size_t)b * QKVN + (size_t)NHEADS * HD + (size_t)g * HD;
  const float* vnew = knew + (size_t)NKV * HD;
  const size_t cbase = ((size_t)b * NKV + g) * CACHELEN * HD;

  const int svalid = start_pos + 1;
  for (int s = wid; s < svalid; s += 8) {
    const float* krow = (s == start_pos) ? knew : (ck + cbase + (size_t)s * HD);
    v4f kf = *(const v4f*)(krow + lane * 4);
    float sc[4];
#pragma unroll
    for (int h = 0; h < 4; ++h) {
      const float* qh = qs + h * HD + lane * 4;
      float p = qh[0] * kf.x + qh[1] * kf.y + qh[2] * kf.z + qh[3] * kf.w;
#pragma unroll
      for (int off = 16; off > 0; off >>= 1) p += __shfl_xor(p, off, 32);
      sc[h] = p;                       // full score, uniform across lanes
    }
    const float* vrow = (s == start_pos) ? vnew : (cv + cbase + (size_t)s * HD);
    v4f vf = *(const v4f*)(vrow + lane * 4);
#pragma unroll
    for (int h = 0; h < 4; ++h) {
      const float mn    = fmaxf(mi[h], sc[h]);
      const float alpha = __expf(mi[h] - mn);
      const float p     = __expf(sc[h] - mn);
      li[h] = li[h] * alpha + p;
      oi[h] = oi[h] * alpha + p * vf;
      mi[h] = mn;
    }
  }

  if (lane == 0) {
#pragma unroll
    for (int h = 0; h < 4; ++h) { m_s[wid][h] = mi[h]; l_s[wid][h] = li[h]; }
  }
#pragma unroll
  for (int h = 0; h < 4; ++h) {
    o_s[wid][h][lane * 4 + 0] = oi[h].x;
    o_s[wid][h][lane * 4 + 1] = oi[h].y;
    o_s[wid][h][lane * 4 + 2] = oi[h].z;
    o_s[wid][h][lane * 4 + 3] = oi[h].w;
  }
  __syncthreads();

  for (int idx = tid; idx < 4 * HD; idx += 256) {
    const int h = idx >> 7;
    const int d = idx & 127;
    float M = -3.0e38f;
#pragma unroll
    for (int w = 0; w < 8; ++w) M = fmaxf(M, m_s[w][h]);
    float L = 0.f, O = 0.f;
#pragma unroll
    for (int w = 0; w < 8; ++w) {
      const float wt = __expf(m_s[w][h] - M);
      L += l_s[w][h] * wt;
      O += o_s[w][h][d] * wt;
    }
    attn[(size_t)b * DIMX + (size_t)g * 4 * HD + idx] = O / L;
  }
}

// ---------------------------------------------------------------------------
extern "C" void kernel_launch(void* const* d_in, const int* in_sizes, int n_in,
                              void* d_out, int out_size, void* d_ws, size_t ws_size,
                              hipStream_t stream) {
  const float* x    = (const float*)d_in[0];   // (32, 4096)
  const float* wqkv = (const float*)d_in[1];   // (4096, 6144)
  const float* wo   = (const float*)d_in[2];   // (4096, 4096)
  const float* ck   = (const float*)d_in[3];   // (32, 8, 2176, 128)
  const float* cv   = (const float*)d_in[4];   // (32, 8, 2176, 128)
  const int*   sp   = (const int*)d_in[5];     // scalar start_pos
  float* out  = (float*)d_out;                 // (32, 4096)

  float* qkv  = (float*)d_ws;                  // 32*6144 f32 = 768 KiB
  float* attn = qkv + (size_t)BATCHSZ * QKVN;  // 32*4096 f32 = 512 KiB

  // 1) fused QKV projection
  gemm32_wmma<<<QKVN / 16, 256, 0, stream>>>(x, wqkv, qkv, DIMX, QKVN);
  // 2) RoPE on q and k rows of qkv
  const int rope_total = BATCHSZ * (NHEADS + NKV) * (HD / 2);
  rope_kernel<<<(rope_total + 255) / 256, 256, 0, stream>>>(qkv, sp);
  // 3) grouped-query flash-decode attention over the cache
  attn_kernel<<<BATCHSZ * NKV, 256, 0, stream>>>(qkv, ck, cv, attn, sp);
  // 4) output projection
  gemm32_wmma<<<DIMX / 16, 256, 0, stream>>>(attn, wo, out, DIMX, DIMX);
}